// MOLAttn_26834955665939
// MI455X (gfx1250) — compile-verified
//
#include <hip/hip_runtime.h>
#include <math.h>

#define B_ 64
#define T_ 2000
#define D_ 512
#define H_ 256
#define P_ 15        // 3*K params per batch
#define TCHUNK 80    // 25 chunks over T

typedef __attribute__((ext_vector_type(2))) float v2f;
typedef __attribute__((ext_vector_type(8))) float v8f;

// ---------------------------------------------------------------------------
// Kernel 1: h = tanh(state @ W1 + b1)   (64x512) @ (512x256) -> (64x256)
// One wave (32 lanes) per 16x16 output tile, V_WMMA_F32_16X16X4_F32, K=512.
// A 16x4 f32 layout (ISA 7.12.2): lanes 0-15 M=0..15 / lanes 16-31 M=0..15;
//   VGPR v holds K = 2*half + v.  B 4x16: row K striped across lanes, same
//   half/K pattern.  C/D: VGPR i -> M = i + 8*half, N = lane & 15.
// ---------------------------------------------------------------------------
__global__ __launch_bounds__(32) void gemm1_wmma(const float* __restrict__ state,
                                                 const float* __restrict__ W1,
                                                 const float* __restrict__ b1,
                                                 float* __restrict__ h) {
    const int mt = blockIdx.x & 3;    // 4 M-tiles  (B = 64)
    const int nt = blockIdx.x >> 2;   // 16 N-tiles (H = 256)
    const int m0 = mt * 16, n0 = nt * 16;
    const int lane = threadIdx.x;
    const int half = lane >> 4;       // 0: lanes 0-15, 1: lanes 16-31
    const int r    = lane & 15;

    v8f acc = {};
    const float* arow = state + (size_t)(m0 + r) * D_;
    for (int k = 0; k < D_; k += 4) {
        const int ka = k + 2 * half;
        v2f a, b;
        a[0] = arow[ka + 0];
        a[1] = arow[ka + 1];
        b[0] = W1[(size_t)(ka + 0) * H_ + n0 + r];
        b[1] = W1[(size_t)(ka + 1) * H_ + n0 + r];
        acc = __builtin_amdgcn_wmma_f32_16x16x4_f32(
            /*neg_a=*/false, a, /*neg_b=*/false, b,
            /*c_mod=*/(short)0, acc, /*reuse_a=*/false, /*reuse_b=*/false);
    }

    const float bias = b1[n0 + r];
#pragma unroll
    for (int i = 0; i < 8; ++i) {
        const int row = m0 + i + 8 * half;   // M
        h[(size_t)row * H_ + n0 + r] = tanhf(acc[i] + bias);
    }
}

// ---------------------------------------------------------------------------
// Kernel 2: params = exp(h @ W2 + b2)   (64x256)@(256x15) -> (64x15)
// One thread per output scalar (960 threads total).
// ---------------------------------------------------------------------------
__global__ void param_kernel(const float* __restrict__ h,
                             const float* __restrict__ W2,
                             const float* __restrict__ b2,
                             float* __restrict__ params) {
    const int idx = blockIdx.x * blockDim.x + threadIdx.x;
    if (idx >= B_ * P_) return;
    const int b = idx / P_, p = idx % P_;
    float s = b2[p];
    const float* hb = h + (size_t)b * H_;
#pragma unroll 4
    for (int k = 0; k < H_; ++k) s += hb[k] * W2[(size_t)k * P_ + p];
    params[idx] = expf(s);
}

// ---------------------------------------------------------------------------
// Kernel 3: align[b,t] = sum_k CDF(t+0.5) - CDF(t-0.5)
// softmax over the singleton axis==1 is identically 1.0, so weights drop out.
// ---------------------------------------------------------------------------
__device__ __forceinline__ float sigmoidf(float x) {
    return 1.0f / (1.0f + expf(-x));
}

__global__ void align_kernel(const float* __restrict__ params,
                             float* __restrict__ align_out) {
    const int idx = blockIdx.x * blockDim.x + threadIdx.x;
    if (idx >= B_ * T_) return;
    const int b = idx / T_;
    const int t = idx % T_;
    const float* p = params + (size_t)b * P_;
    const float tj = (float)t;
    float s = 0.0f;
#pragma unroll
    for (int k = 0; k < 5; ++k) {
        const float mean  = p[3 * k + 0];
        const float inv_s = 1.0f / p[3 * k + 1];
        s += sigmoidf((tj + 0.5f - mean) * inv_s)
           - sigmoidf((tj - 0.5f - mean) * inv_s);
    }
    align_out[idx] = s;
}

// ---------------------------------------------------------------------------
// Kernel 4: zero d_out (harness poisons it; context kernel accumulates).
// ---------------------------------------------------------------------------
__global__ void zero_kernel(float* __restrict__ out, int n) {
    const int idx = blockIdx.x * blockDim.x + threadIdx.x;
    if (idx < n) out[idx] = 0.0f;
}

// ---------------------------------------------------------------------------
// Kernel 5: context[b,d] = sum_t align[b,t] * enc_z[b,t,d]
// The bandwidth kernel: 262 MB of enc_z streamed once.  grid = (64 batches,
// 25 T-chunks) x 128 threads (4 waves); each thread owns one float4 of D.
// b128 loads, prefetch 4 rows ahead, f32 global atomics to combine chunks.
// ---------------------------------------------------------------------------
__global__ __launch_bounds__(128) void context_kernel(const float* __restrict__ enc_z,
                                                      const float* __restrict__ align_in,
                                                      float* __restrict__ out) {
    const int b  = blockIdx.x;
    const int t0 = blockIdx.y * TCHUNK;
    const int d4 = threadIdx.x;                       // 0..127 -> 4 floats each
    const float4* ez = (const float4*)(enc_z + (size_t)b * T_ * D_);
    const float* al  = align_in + (size_t)b * T_;

    float4 acc = make_float4(0.f, 0.f, 0.f, 0.f);
#pragma unroll 4
    for (int t = t0; t < t0 + TCHUNK; ++t) {
        // prefetch 4 rows ahead (clamped inside the tensor) -> global_prefetch_b8
        const int tp = (t + 4 < T_) ? (t + 4) : (T_ - 1);
        __builtin_prefetch(&ez[(size_t)tp * (D_ / 4) + d4], 0, 0);

        const float  w = al[t];
        const float4 v = ez[(size_t)t * (D_ / 4) + d4];
        acc.x = fmaf(w, v.x, acc.x);
        acc.y = fmaf(w, v.y, acc.y);
        acc.z = fmaf(w, v.z, acc.z);
        acc.w = fmaf(w, v.w, acc.w);
    }

    float* o = out + (size_t)b * D_ + d4 * 4;
    atomicAdd(o + 0, acc.x);
    atomicAdd(o + 1, acc.y);
    atomicAdd(o + 2, acc.z);
    atomicAdd(o + 3, acc.w);
}

// ---------------------------------------------------------------------------
extern "C" void kernel_launch(void* const* d_in, const int* in_sizes, int n_in,
                              void* d_out, int out_size, void* d_ws, size_t ws_size,
                              hipStream_t stream) {
    const float* state = (const float*)d_in[0];   // (64,1,512)
    const float* enc_z = (const float*)d_in[1];   // (64,2000,512)
    const float* W1    = (const float*)d_in[2];   // (512,256)
    const float* b1    = (const float*)d_in[3];   // (256,)
    const float* W2    = (const float*)d_in[4];   // (256,15)
    const float* b2    = (const float*)d_in[5];   // (15,)
    float* out = (float*)d_out;                   // (64,1,512)

    float* ws      = (float*)d_ws;
    float* h       = ws;               // 16384 floats
    float* params  = ws + 16384;       // 960 floats
    float* align_b = ws + 17408;       // 128000 floats  (~582 KB total used)

    gemm1_wmma   <<<dim3(64),                    dim3(32),  0, stream>>>(state, W1, b1, h);
    param_kernel <<<dim3((B_ * P_ + 255) / 256), dim3(256), 0, stream>>>(h, W2, b2, params);
    align_kernel <<<dim3((B_ * T_ + 255) / 256), dim3(256), 0, stream>>>(params, align_b);
    zero_kernel  <<<dim3((B_ * D_ + 255) / 256), dim3(256), 0, stream>>>(out, B_ * D_);
    context_kernel<<<dim3(B_, T_ / TCHUNK),      dim3(128), 0, stream>>>(enc_z, align_b, out);
}